// LaneNetEmbedingLoss_86011015070366
// MI455X (gfx1250) — compile-verified
//
#include <hip/hip_runtime.h>
#include <hip/hip_bf16.h>
#include <math.h>

typedef float v2f __attribute__((ext_vector_type(2)));
typedef float v8f __attribute__((ext_vector_type(8)));

#define C_CLASSES 5
#define D_DIM     8
#define B_SAMP    8
#define ACCW      10               // [count, sum0..sum7, sumsq]
#define NACC      (C_CLASSES * ACCW)   // 50
#define BPS       64               // blocks per sample

// d_ws layout (floats):
//   [0, B*NACC*BPS)            partials, transposed: partials[(b*NACC + t)*BPS + blk]
//   [B*NACC*BPS, +B*NACC)      folded totals: tot[b*NACC + t]
#define WS_PARTIALS 0
#define WS_TOTALS   (B_SAMP * NACC * BPS)

__device__ __forceinline__ float wave_reduce_add(float v) {
#pragma unroll
  for (int off = 16; off > 0; off >>= 1)
    v += __shfl_xor(v, off, 32);
  return v;
}

// ---------------------------------------------------------------------------
// Stage 1: single streaming pass over emb/seg (the HBM-bound 288 MB).
// Per (sample, block): 50-float partial written to d_ws (transposed layout).
// ---------------------------------------------------------------------------
__global__ void lanenet_partial_kernel(const float* __restrict__ emb,
                                       const int* __restrict__ seg,
                                       float* __restrict__ ws,
                                       int quads) {
  const int b    = blockIdx.y;
  const int nthr = blockDim.x * gridDim.x;
  const int tid0 = blockIdx.x * blockDim.x + threadIdx.x;

  const int4* __restrict__ seg4 =
      reinterpret_cast<const int4*>(seg) + (size_t)b * quads;
  const float4* __restrict__ ch[D_DIM];
#pragma unroll
  for (int d = 0; d < D_DIM; ++d)
    ch[d] = reinterpret_cast<const float4*>(emb) + ((size_t)(b * D_DIM + d)) * quads;

  float acc[C_CLASSES][ACCW];
#pragma unroll
  for (int c = 0; c < C_CLASSES; ++c)
#pragma unroll
    for (int k = 0; k < ACCW; ++k) acc[c][k] = 0.0f;

  for (int q = tid0; q < quads; q += nthr) {
    const int qn = q + nthr;
    __builtin_prefetch(seg4 + (qn < quads ? qn : q), 0, 3);

    const int4 s4 = seg4[q];
    float f[4][D_DIM];
#pragma unroll
    for (int d = 0; d < D_DIM; ++d) {
      const float4 v = ch[d][q];
      f[0][d] = v.x; f[1][d] = v.y; f[2][d] = v.z; f[3][d] = v.w;
    }
    const int sj[4] = {s4.x, s4.y, s4.z, s4.w};
#pragma unroll
    for (int j = 0; j < 4; ++j) {
      float sq = 0.0f;
#pragma unroll
      for (int d = 0; d < D_DIM; ++d) sq = fmaf(f[j][d], f[j][d], sq);
#pragma unroll
      for (int c = 0; c < C_CLASSES; ++c) {
        const float m = (sj[j] == c) ? 1.0f : 0.0f;   // one-hot, predicated
        acc[c][0] += m;
#pragma unroll
        for (int d = 0; d < D_DIM; ++d)
          acc[c][1 + d] = fmaf(m, f[j][d], acc[c][1 + d]);
        acc[c][9] = fmaf(m, sq, acc[c][9]);
      }
    }
  }

  __shared__ float lds[8][NACC];
  const int lane = threadIdx.x & 31;
  const int wv   = threadIdx.x >> 5;
#pragma unroll
  for (int c = 0; c < C_CLASSES; ++c)
#pragma unroll
    for (int k = 0; k < ACCW; ++k) {
      const float v = wave_reduce_add(acc[c][k]);
      if (lane == 0) lds[wv][c * ACCW + k] = v;
    }
  __syncthreads();
  const int nwaves = blockDim.x >> 5;
  if (threadIdx.x < NACC) {
    float t = 0.0f;
    for (int w = 0; w < nwaves; ++w) t += lds[w][threadIdx.x];
    // transposed: consecutive blk for fixed (b, t) -> coalesced fold loads
    ws[WS_PARTIALS + ((size_t)b * NACC + threadIdx.x) * BPS + blockIdx.x] = t;
  }
}

// ---------------------------------------------------------------------------
// Stage 2: fold the BPS block-partials lane-parallel (coalesced, pipelined).
// Grid = B samples, block = 256 (8 waves); wave w handles t = w, w+8, ...
// ---------------------------------------------------------------------------
__global__ void lanenet_fold_kernel(float* __restrict__ ws) {
  const int b    = blockIdx.x;
  const int lane = threadIdx.x & 31;
  const int wv   = threadIdx.x >> 5;
  for (int t = wv; t < NACC; t += 8) {
    const float* p = ws + WS_PARTIALS + ((size_t)b * NACC + t) * BPS;
    float s = p[lane] + p[lane + 32];           // BPS = 64 = 2 * wave32
    s = wave_reduce_add(s);                     // fixed-order -> deterministic
    if (lane == 0) ws[WS_TOTALS + b * NACC + t] = s;
  }
}

// ---------------------------------------------------------------------------
// Stage 3: one wave32. Means, Gram = M*M^T via V_WMMA_F32_16X16X4_F32
// (two K=4 steps over D=8), hinge + var losses, scalar output. Branchless
// operand construction (clamped indices + cndmask masks) to keep EXEC full
// and avoid exec-masked ds_load chains.
// ---------------------------------------------------------------------------
__global__ void lanenet_finalize_kernel(const float* __restrict__ ws,
                                        float* __restrict__ out) {
  __shared__ float tot[NACC];
  const int lane = threadIdx.x;          // blockDim.x == 32, EXEC all ones
  float loss = 0.0f;

  for (int b = 0; b < B_SAMP; ++b) {
    for (int t = lane; t < NACC; t += 32)
      tot[t] = ws[WS_TOTALS + b * NACC + t];
    __syncthreads();

    // A operand for 16x4 f32 WMMA: lane (row = L&15, half h = L>>4) holds
    // {mean[row][kb+2h], mean[row][kb+2h+1]}. Rows >= C zero-padded.
    // Gram is symmetric and B(4x16) mirrors A, so A doubles as B.
    const int   row   = lane & 15;
    const int   h     = lane >> 4;
    const bool  valid = row < C_CLASSES;
    const int   r     = valid ? row : 0;          // clamped -> unconditional LDS loads
    const float vm    = valid ? 1.0f : 0.0f;
    const float inv   = vm / tot[r * ACCW];
    v2f a0, a1;
    a0.x = tot[r * ACCW + 1 + 2 * h + 0] * inv;   // K = 0+2h
    a0.y = tot[r * ACCW + 1 + 2 * h + 1] * inv;   // K = 1+2h
    a1.x = tot[r * ACCW + 5 + 2 * h + 0] * inv;   // K = 4+2h
    a1.y = tot[r * ACCW + 5 + 2 * h + 1] * inv;   // K = 5+2h

    v8f g = {};
    g = __builtin_amdgcn_wmma_f32_16x16x4_f32(false, a0, false, a0, (short)0, g, false, false);
    g = __builtin_amdgcn_wmma_f32_16x16x4_f32(false, a1, false, a1, (short)0, g, false, false);

    // lane j (j<16) holds column j: gc[i] = G[i][j] for rows i=0..4 (VGPRs 0..4)
    float gc[C_CLASSES] = {g[0], g[1], g[2], g[3], g[4]};
    float diag[C_CLASSES];
#pragma unroll
    for (int i = 0; i < C_CLASSES; ++i) diag[i] = __shfl(gc[i], i, 32);

    const bool  jv = lane < C_CLASSES;
    const float diag_j = (lane == 0) ? gc[0] : (lane == 1) ? gc[1]
                       : (lane == 2) ? gc[2] : (lane == 3) ? gc[3] : gc[4];
    float hinge = 0.0f;
#pragma unroll
    for (int i = 0; i < C_CLASSES; ++i) {
      const float d2 = diag[i] + diag_j - 2.0f * gc[i];
      const float d  = sqrtf(fmaxf(d2, 0.0f));
      const float hv = fmaxf(3.0f - d, 0.0f);
      hinge += (jv && (i != lane)) ? hv : 0.0f;   // cndmask, no divergence
    }

    // var loss via ss_c = sumsq_c - count_c * ||mean_c||^2 (single-pass identity)
    const float cnt = tot[r * ACCW];
    const float ic  = 1.0f / cnt;
    float m2 = 0.0f;
#pragma unroll
    for (int d = 0; d < D_DIM; ++d) {
      const float mm = tot[r * ACCW + 1 + d] * ic;
      m2 = fmaf(mm, mm, m2);
    }
    const float ss    = tot[r * ACCW + 9] - cnt * m2;
    const float var_t = jv ? fmaxf(sqrtf(fmaxf(ss, 0.0f)) - 0.5f, 0.0f) : 0.0f;

    loss += wave_reduce_add(var_t) * (1.0f / C_CLASSES)
          + wave_reduce_add(hinge) * (1.0f / (C_CLASSES * (C_CLASSES - 1)));
    __syncthreads();  // protect tot[] before next sample's writes
  }

  if (lane == 0) out[0] = loss * (1.0f / B_SAMP);
}

extern "C" void kernel_launch(void* const* d_in, const int* in_sizes, int n_in,
                              void* d_out, int out_size, void* d_ws, size_t ws_size,
                              hipStream_t stream) {
  (void)n_in; (void)out_size; (void)ws_size;
  const float* emb = (const float*)d_in[0];
  const int*   seg = (const int*)d_in[1];
  float* out       = (float*)d_out;
  float* ws        = (float*)d_ws;   // ~102 KB used, fully overwritten each call

  const int HW    = in_sizes[1] / B_SAMP;  // pixels per sample
  const int quads = HW / 4;

  dim3 grid(BPS, B_SAMP);
  lanenet_partial_kernel<<<grid, 256, 0, stream>>>(emb, seg, ws, quads);
  lanenet_fold_kernel<<<B_SAMP, 256, 0, stream>>>(ws);
  lanenet_finalize_kernel<<<1, 32, 0, stream>>>(ws, out);
}